// RewardFunctionRegret_70549132804459
// MI455X (gfx1250) — compile-verified
//
#include <hip/hip_runtime.h>
#include <hip/hip_bf16.h>

typedef float v2f __attribute__((ext_vector_type(2)));
typedef float v8f __attribute__((ext_vector_type(8)));

#define Gdim 128
#define Pdim 64
#define NFdim 6
#define INV_T 1000.0f   // 1/T, T = 0.001
#define LDS_STRIDE 68   // 64 + 4 pad -> conflict-free column access

// Extract element i (0..7) of a v8f accumulator without scratch spill.
__device__ __forceinline__ float vsel8(v8f c, int i) {
  float r = c[0];
  r = (i == 1) ? c[1] : r;
  r = (i == 2) ? c[2] : r;
  r = (i == 3) ? c[3] : r;
  r = (i == 4) ? c[4] : r;
  r = (i == 5) ? c[5] : r;
  r = (i == 6) ? c[6] : r;
  r = (i == 7) ? c[7] : r;
  return r;
}

// Phase 1: one block per (x, 16-wide y chunk). 8 waves; wave w covers p = 8w..8w+7.
// Each (wave, p) does a 16-row x K=6 dot-with-W via two V_WMMA_F32_16X16X4_F32
// (K padded to 8; B = W broadcast across all 16 columns). Results land in LDS
// as vs[cell][p]; 16 threads then do the T=0.001 softmax-weighted sum per cell.
__global__ void __launch_bounds__(256)
rfr_phase1_valtable(const float* __restrict__ succ, const float* __restrict__ Wg,
                    float* __restrict__ val) {
  __shared__ float vs_lds[16 * LDS_STRIDE];

  const int bid  = blockIdx.x;          // 0 .. 128*8-1
  const int x    = bid >> 3;
  const int y0   = (bid & 7) << 4;
  const int tid  = threadIdx.x;
  const int wave = tid >> 5;            // 0..7
  const int lane = tid & 31;
  const int m    = lane & 15;           // row within 16-row tile
  const int hi   = lane >> 4;           // half-wave: K 2/3 side of A, rows 2/3 of B

  const float w0 = Wg[0], w1 = Wg[1], w2 = Wg[2];
  const float w3 = Wg[3], w4 = Wg[4], w5 = Wg[5];

  // B-matrix (4x16) per ISA layout: VGPR0 lanes0-15=row0, lanes16-31=row2;
  // VGPR1 lanes0-15=row1, lanes16-31=row3. Every column identical = W[k].
  v2f b1; b1.x = hi ? w2  : w0;  b1.y = hi ? w3  : w1;   // K = 0..3
  v2f b2; b2.x = hi ? 0.f : w4;  b2.y = hi ? 0.f : w5;   // K = 4..7 (6,7 = 0)

  #pragma unroll
  for (int pi = 0; pi < 8; ++pi) {
    const int p = wave * 8 + pi;
    const float* base = succ + (size_t)((p * Gdim + x) * Gdim + y0) * NFdim;
    const float* rowp = base + m * NFdim;               // 8B-aligned (m*24)

    // A-matrix (16x4) per ISA layout: lanes0-15 VGPR0/1 = K0/K1,
    // lanes16-31 VGPR0/1 = K2/K3. Streamed once -> non-temporal loads.
    v2f a1 = __builtin_nontemporal_load((const v2f*)(rowp + (hi ? 2 : 0)));
    v2f a2;
    if (hi) { a2.x = 0.f; a2.y = 0.f; }
    else    { a2 = __builtin_nontemporal_load((const v2f*)(rowp + 4)); }

    v8f c = {};
    c = __builtin_amdgcn_wmma_f32_16x16x4_f32(false, a1, false, b1, (short)0, c, false, false);
    c = __builtin_amdgcn_wmma_f32_16x16x4_f32(false, a2, false, b2, (short)0, c, false, false);

    // D[m][n] = dot_m for every n. Lane L<16 VGPR r -> dot_r; lane>=16 -> dot_{8+r}.
    // Lanes 0-7 emit rows 0-7, lanes 16-23 emit rows 8-15.
    const int r = lane & 7;
    if ((lane < 8) || (lane >= 16 && lane < 24)) {
      const int row = (lane >> 4) * 8 + r;
      vs_lds[row * LDS_STRIDE + p] = vsel8(c, r);
    }
  }
  __syncthreads();

  if (tid < 16) {
    const float* col = &vs_lds[tid * LDS_STRIDE];
    float mx = -INFINITY;
    #pragma unroll 8
    for (int p = 0; p < Pdim; ++p) mx = fmaxf(mx, col[p]);
    float se = 0.f, sw = 0.f;
    #pragma unroll 8
    for (int p = 0; p < Pdim; ++p) {
      const float v = col[p];
      const float e = __expf((v - mx) * INV_T);
      se += e;
      sw += e * v;
    }
    val[x * Gdim + y0 + tid] = sw / se;
  }
}

// Phase 2: thread per batch element. phi row = 20 contiguous floats
// ([t=0: 6 feats + 4 coords][t=1: ...]). val gathers hit L2 (64 KB table).
__global__ void __launch_bounds__(256)
rfr_phase2_out(const float* __restrict__ phi, const float* __restrict__ Wg,
               const float* __restrict__ val, float* __restrict__ out, int B) {
  const int b = blockIdx.x * blockDim.x + threadIdx.x;
  if (b >= B) return;

  float w[NFdim];
  #pragma unroll
  for (int f = 0; f < NFdim; ++f) w[f] = Wg[f];

  const float* ph = phi + (size_t)b * 20;
  float der[2];
  #pragma unroll
  for (int t = 0; t < 2; ++t) {
    const float* pt = ph + t * 10;
    float pr = 0.f;
    #pragma unroll
    for (int f = 0; f < NFdim; ++f) pr = fmaf(pt[f], w[f], pr);
    const int xs = (int)pt[6], ys = (int)pt[7];
    const int xe = (int)pt[8], ye = (int)pt[9];
    der[t] = pr + val[xe * Gdim + ye] - val[xs * Gdim + ys];
  }
  const float d = der[0] - der[1];
  out[b * 2 + 0] = 1.f / (1.f + __expf(-d));   // sigmoid(d)
  out[b * 2 + 1] = 1.f / (1.f + __expf(d));    // sigmoid(-d)
}

extern "C" void kernel_launch(void* const* d_in, const int* in_sizes, int n_in,
                              void* d_out, int out_size, void* d_ws, size_t ws_size,
                              hipStream_t stream) {
  const float* phi  = (const float*)d_in[0];   // (B, 2, 10)
  const float* succ = (const float*)d_in[1];   // (64, 128, 128, 6)
  const float* W    = (const float*)d_in[2];   // (6,)
  float* out = (float*)d_out;                  // (B, 2, 1)
  float* val = (float*)d_ws;                   // 128*128 floats = 64 KB

  const int B = in_sizes[0] / 20;

  // Phase 1: 128 x-rows * 8 y-chunks = 1024 blocks, 256 threads (8 waves).
  rfr_phase1_valtable<<<Gdim * (Gdim / 16), 256, 0, stream>>>(succ, W, val);

  // Phase 2: thread per batch element.
  rfr_phase2_out<<<(B + 255) / 256, 256, 0, stream>>>(phi, W, val, out, B);
}